// DRRAA_nre_28235115004162
// MI455X (gfx1250) — compile-verified
//
#include <hip/hip_runtime.h>
#include <math.h>

#define NN 100000
#define KK 8
#define DD 2
#define SS 5000
#define EE 400000
#define EPSF 1e-6f

typedef __attribute__((ext_vector_type(2))) float v2f;
typedef __attribute__((ext_vector_type(8))) float v8f;

// ---- workspace layout (float offsets) ----
#define WS_ZS     0                       // Zs  (K,N) col-softmax   : 800000
#define WS_ZG     (WS_ZS + KK*NN)         // ZG  (N,K)               : 800000
#define WS_EMB    (WS_ZG + NN*KK)         // EmbAll (N,2)            : 200000
#define WS_SX     (WS_EMB + 2*NN)         // sample emb x            : 5000
#define WS_SY     (WS_SX + SS)            // sample emb y            : 5000
#define WS_USQ    (WS_SY + SS)            // |x+eps|^2               : 5000
#define WS_XSQ    (WS_USQ + SS)           // |x|^2                   : 5000
#define WS_BS     (WS_XSQ + SS)           // beta at samples         : 5000
#define WS_COLSUM (WS_BS + SS)            // 8
#define WS_M16    (WS_COLSUM + 8)         // 16x16 gram (8x8 used)   : 256
#define WS_AZC    (WS_M16 + 256)          // 2x8                     : 16
#define WS_ACC    (WS_AZC + 16)           // pairSum, diagSum, edgeSum, spare
#define WS_ZERO_BEGIN WS_COLSUM
#define WS_ZERO_COUNT (8 + 256 + 16 + 4)

#define TI  ((SS + 15) / 16)              // 313 j/i tiles per dim
#define JT  8                             // j-tiles per wave strip
#define TJS ((TI + JT - 1) / JT)          // 40 strips per tile row

__device__ __forceinline__ float waveReduce(float v) {
#pragma unroll
    for (int o = 16; o > 0; o >>= 1) v += __shfl_xor(v, o, 32);
    return v;
}

// ---- kernel 0: zero accumulators ----
__global__ void k_init(float* ws) {
    int t = blockIdx.x * blockDim.x + threadIdx.x;
    if (t < WS_ZERO_COUNT) ws[WS_ZERO_BEGIN + t] = 0.0f;
}

// ---- kernel 1: softmax over K, sigmoid gate, ZG, column sums ----
__global__ void k_softmax_zg(const float* __restrict__ Z,
                             const float* __restrict__ Gate,
                             float* __restrict__ ws) {
    __shared__ float red[8][KK];
    int n = blockIdx.x * blockDim.x + threadIdx.x;
    int wave = threadIdx.x >> 5;
    int lane = threadIdx.x & 31;
    bool act = n < NN;
    float zg[KK] = {0.f, 0.f, 0.f, 0.f, 0.f, 0.f, 0.f, 0.f};
    if (act) {
        float z[KK];
        float m = -1e30f;
#pragma unroll
        for (int k = 0; k < KK; k++) { z[k] = Z[k * NN + n]; m = fmaxf(m, z[k]); }
        float s = 0.f;
#pragma unroll
        for (int k = 0; k < KK; k++) { z[k] = expf(z[k] - m); s += z[k]; }
        float inv = 1.f / s;
#pragma unroll
        for (int k = 0; k < KK; k++) {
            float zs = z[k] * inv;
            ws[WS_ZS + k * NN + n] = zs;
            float g = 1.f / (1.f + expf(-Gate[n * KK + k]));
            zg[k] = zs * g;
            ws[WS_ZG + n * KK + k] = zg[k];
        }
    }
#pragma unroll
    for (int k = 0; k < KK; k++) {
        float v = waveReduce(zg[k]);
        if (lane == 0) red[wave][k] = v;
    }
    __syncthreads();
    if (threadIdx.x < KK) {
        float s = 0.f;
#pragma unroll
        for (int w = 0; w < 8; w++) s += red[w][threadIdx.x];
        atomicAdd(&ws[WS_COLSUM + threadIdx.x], s);
    }
}

// ---- kernel 2: M = Z_samp @ C_samp via V_WMMA_F32_16X16X4_F32 ----
// A tile 16x4: rows = Zs component index (0..7 live), K slots = 4 samples.
// B tile 4x16: cols = C component index (0..7 live).
// Unconditional clamped loads + multiplicative masks (no EXEC divergence).
__global__ void k_gram8(const int* __restrict__ idx, float* __restrict__ ws) {
    const float* Zs = ws + WS_ZS;
    const float* ZG = ws + WS_ZG;
    int lane = threadIdx.x & 31;
    int wave = threadIdx.x >> 5;
    int low = lane & 15;   // A row index / B col index
    int hi  = lane >> 4;   // selects K slots {2*hi, 2*hi+1}
    float rowMask = (low < KK) ? 1.f : 0.f;
    float invc = (low < KK) ? (1.f / ws[WS_COLSUM + low]) : 0.f;
    v8f c = {};
    for (int s0 = wave * 4; s0 < SS; s0 += 32) {
        int sA = s0 + 2 * hi;
        int sB = sA + 1;
        int iA = idx[min(sA, SS - 1)];
        int iB = idx[min(sB, SS - 1)];
        float mA = (sA < SS) ? rowMask : 0.f;
        float mB = (sB < SS) ? rowMask : 0.f;
        v2f a, b;
        a.x = mA * Zs[low * NN + iA];                 // rows 8..15 read padded ws, masked
        a.y = mB * Zs[low * NN + iB];
        b.x = mA * ZG[iA * KK + low] * invc;
        b.y = mB * ZG[iB * KK + low] * invc;
        c = __builtin_amdgcn_wmma_f32_16x16x4_f32(false, a, false, b,
                                                  (short)0, c, false, false);
    }
    int m0 = 8 * hi;
#pragma unroll
    for (int r = 0; r < 8; r++)
        atomicAdd(&ws[WS_M16 + (m0 + r) * 16 + low], c[r]);
}

// ---- kernel 3a: AZC = A @ M (2x8) ----
__global__ void k_azc(const float* __restrict__ A, float* __restrict__ ws) {
    int t = threadIdx.x;
    if (t < DD * KK) {
        int d = t >> 3, k = t & 7;
        float s = 0.f;
#pragma unroll
        for (int m = 0; m < KK; m++) s += A[d * KK + m] * ws[WS_M16 + m * 16 + k];
        ws[WS_AZC + d * KK + k] = s;
    }
}

// ---- kernel 3b: sample embeddings, norms, beta gather, diagonal sum ----
__global__ void k_samples(const int* __restrict__ sidx,
                          const float* __restrict__ beta,
                          float* __restrict__ ws) {
    int s = blockIdx.x * blockDim.x + threadIdx.x;
    float diag = 0.f;
    if (s < SS) {
        int i = sidx[s];
        float x = 0.f, y = 0.f;
#pragma unroll
        for (int k = 0; k < KK; k++) {
            float zv = ws[WS_ZS + k * NN + i];
            x += ws[WS_AZC + k] * zv;
            y += ws[WS_AZC + KK + k] * zv;
        }
        ws[WS_SX + s] = x;
        ws[WS_SY + s] = y;
        float ux = x + EPSF, uy = y + EPSF;
        ws[WS_USQ + s] = ux * ux + uy * uy;
        ws[WS_XSQ + s] = x * x + y * y;
        float b = beta[i];
        ws[WS_BS + s] = b;
        diag = expf(2.f * b - sqrtf(2.f * EPSF * EPSF));   // dist_ii = eps*sqrt(2)
    }
    float v = waveReduce(diag);
    if ((threadIdx.x & 31) == 0) atomicAdd(&ws[WS_ACC + 1], v);
}

// ---- kernel 4: S x S pair sum; strip of 8 j-tiles per wave, A-frag reused ----
__global__ void k_pairs(float* __restrict__ ws) {
    const float* sx  = ws + WS_SX;
    const float* sy  = ws + WS_SY;
    const float* usq = ws + WS_USQ;
    const float* xsq = ws + WS_XSQ;
    const float* bs  = ws + WS_BS;
    __shared__ float red[8];
    int wave = threadIdx.x >> 5;
    int lane = threadIdx.x & 31;
    int low = lane & 15, hi = lane >> 4;
    float sum = 0.f;
    long t = (long)blockIdx.x * 8 + wave;
    if (t < (long)TI * TJS) {                       // wave-uniform
        int ti = (int)(t / TJS);
        int js = (int)(t % TJS);
        int i0 = ti * 16;
        // A: u_i = x_i + eps (16x4, K0=ux K1=uy, K2=K3=0)
        int ia  = i0 + low;
        int iac = min(ia, SS - 1);
        float mA = (hi == 0 && ia < SS) ? 1.f : 0.f;
        v2f a;
        a.x = mA * (sx[iac] + EPSF);
        a.y = mA * (sy[iac] + EPSF);
        // i-side epilogue data, loaded once per strip
        float usqi[8], bsi[8];
        bool  vi[8];
#pragma unroll
        for (int r = 0; r < 8; r++) {
            int ii  = i0 + r + 8 * hi;
            int iic = min(ii, SS - 1);
            vi[r]   = ii < SS;
            usqi[r] = usq[iic];
            bsi[r]  = bs[iic];
        }
        for (int jt = 0; jt < JT; jt++) {
            int tj = js * JT + jt;
            if (tj >= TI) break;                    // wave-uniform
            int j0 = tj * 16;
            int j  = j0 + low;
            int jc = min(j, SS - 1);
            float mB = (hi == 0 && j < SS) ? 1.f : 0.f;
            v2f b;
            b.x = mB * sx[jc];
            b.y = mB * sy[jc];
            v8f c = {};
            c = __builtin_amdgcn_wmma_f32_16x16x4_f32(false, a, false, b,
                                                      (short)0, c, false, false);
            bool  vj  = j < SS;
            float bj  = bs[jc];
            float xj2 = xsq[jc];
#pragma unroll
            for (int r = 0; r < 8; r++) {
                if (vj && vi[r]) {
                    float d2 = fmaxf(usqi[r] + xj2 - 2.f * c[r], 0.f);
                    sum += expf(bsi[r] + bj - sqrtf(d2));
                }
            }
        }
    }
    sum = waveReduce(sum);
    if (lane == 0) red[wave] = sum;
    __syncthreads();
    if (threadIdx.x == 0) {
        float s = 0.f;
#pragma unroll
        for (int w = 0; w < 8; w++) s += red[w];
        atomicAdd(&ws[WS_ACC + 0], s);
    }
}

// ---- kernel 5: embeddings for all N nodes ----
__global__ void k_emb_all(float* __restrict__ ws) {
    int n = blockIdx.x * blockDim.x + threadIdx.x;
    if (n >= NN) return;
    float x = 0.f, y = 0.f;
#pragma unroll
    for (int k = 0; k < KK; k++) {
        float zv = ws[WS_ZS + k * NN + n];
        x += ws[WS_AZC + k] * zv;
        y += ws[WS_AZC + KK + k] * zv;
    }
    ws[WS_EMB + 2 * n]     = x;
    ws[WS_EMB + 2 * n + 1] = y;
}

// ---- kernel 6: edge sum ----
__global__ void k_edges(const int* __restrict__ si, const int* __restrict__ sj,
                        const float* __restrict__ beta, float* __restrict__ ws) {
    __shared__ float red[8];
    int e = blockIdx.x * blockDim.x + threadIdx.x;
    int wave = threadIdx.x >> 5;
    int lane = threadIdx.x & 31;
    float contrib = 0.f;
    if (e < EE) {
        int i = si[e], j = sj[e];
        float dx = ws[WS_EMB + 2 * i]     - ws[WS_EMB + 2 * j]     + EPSF;
        float dy = ws[WS_EMB + 2 * i + 1] - ws[WS_EMB + 2 * j + 1] + EPSF;
        contrib = beta[i] + beta[j] - sqrtf(dx * dx + dy * dy);
    }
    float v = waveReduce(contrib);
    if (lane == 0) red[wave] = v;
    __syncthreads();
    if (threadIdx.x == 0) {
        float s = 0.f;
#pragma unroll
        for (int w = 0; w < 8; w++) s += red[w];
        atomicAdd(&ws[WS_ACC + 2], s);
    }
}

// ---- kernel 7: combine ----
__global__ void k_final(const float* __restrict__ ws, float* __restrict__ out) {
    if (threadIdx.x == 0 && blockIdx.x == 0) {
        float pair = ws[WS_ACC + 0];
        float diag = ws[WS_ACC + 1];
        float edge = ws[WS_ACC + 2];
        out[0] = edge - 0.5f * expf(2.0f) * (pair - diag);
    }
}

extern "C" void kernel_launch(void* const* d_in, const int* in_sizes, int n_in,
                              void* d_out, int out_size, void* d_ws, size_t ws_size,
                              hipStream_t stream) {
    const float* A    = (const float*)d_in[0];
    const float* Z    = (const float*)d_in[1];
    const float* Gate = (const float*)d_in[2];
    const float* beta = (const float*)d_in[3];
    const int*   sidx = (const int*)d_in[4];
    const int*   si   = (const int*)d_in[5];
    const int*   sj   = (const int*)d_in[6];
    float* ws  = (float*)d_ws;
    float* out = (float*)d_out;

    k_init<<<1, 512, 0, stream>>>(ws);
    k_softmax_zg<<<(NN + 255) / 256, 256, 0, stream>>>(Z, Gate, ws);
    k_gram8<<<1, 256, 0, stream>>>(sidx, ws);
    k_azc<<<1, 32, 0, stream>>>(A, ws);
    k_samples<<<(SS + 255) / 256, 256, 0, stream>>>(sidx, beta, ws);
    int waveTasks = TI * TJS;                       // 12520 strip tasks
    k_pairs<<<(waveTasks + 7) / 8, 256, 0, stream>>>(ws);
    k_emb_all<<<(NN + 255) / 256, 256, 0, stream>>>(ws);
    k_edges<<<(EE + 255) / 256, 256, 0, stream>>>(si, sj, beta, ws);
    k_final<<<1, 32, 0, stream>>>(ws, out);
}